// GRAPH_VAE_V3_33148557590873
// MI455X (gfx1250) — compile-verified
//
#include <hip/hip_runtime.h>
#include <math.h>

// ---------------------------------------------------------------------------
// GRAPH_VAE_V3 for gfx1250 (MI455X).
// f32 GEMMs via v_wmma_f32_16x16x4_f32; scatter aggregation via
// global_atomic_add_f32 (unsafeAtomicAdd); BN stats via partial reductions.
// ---------------------------------------------------------------------------

typedef float v2f __attribute__((ext_vector_type(2)));
typedef float v8f __attribute__((ext_vector_type(8)));

#define TPB 256

// ---------------- degree / normalization ----------------
__global__ void deg_accum_kernel(const int* __restrict__ dst, float* __restrict__ deg, int E) {
    int e = blockIdx.x * blockDim.x + threadIdx.x;
    if (e >= E) return;
    unsafeAtomicAdd(&deg[dst[e]], 1.0f);
}

__global__ void deg_to_dinv_kernel(float* __restrict__ deg, int n) {
    int i = blockIdx.x * blockDim.x + threadIdx.x;
    if (i >= n) return;
    deg[i] = rsqrtf(deg[i] + 1.0f);   // self-loop guarantees deg >= 1
}

__global__ void edge_norm_kernel(const int* __restrict__ src, const int* __restrict__ dst,
                                 const float* __restrict__ dinv, float* __restrict__ norm, int E) {
    int e = blockIdx.x * blockDim.x + threadIdx.x;
    if (e >= E) return;
    norm[e] = dinv[src[e]] * dinv[dst[e]];
}

// ---------------- WMMA f32 GEMM: C[nrows,O] = A[nrows,K] @ W[K,O] ----------
// One wave computes one 16x16 output tile. nrows % 16 == 0, K % 4 == 0,
// O % 16 == 0 (holds: 50000 = 3125*16; K in {64,128}; O in {64,128,32}).
__global__ __launch_bounds__(128)
void gemm_wmma_kernel(const float* __restrict__ A, const float* __restrict__ W,
                      float* __restrict__ C, int nrows, int K, int O) {
    const int lane  = threadIdx.x & 31;
    const int wave  = threadIdx.x >> 5;
    const int wpb   = blockDim.x >> 5;
    const int tilesN = O >> 4;
    const int tilesM = nrows >> 4;
    int wid = blockIdx.x * wpb + wave;            // wave-uniform
    if (wid >= tilesM * tilesN) return;           // whole wave exits together
    const int mt = wid / tilesN;
    const int nt = wid - mt * tilesN;
    const int m0 = mt << 4, n0 = nt << 4;

    const int half = lane >> 4;                   // 0: K-pair {0,1}; 1: {2,3}
    const int l15  = lane & 15;
    const int arow = m0 + l15;                    // A row for this lane
    const int bcol = n0 + l15;                    // B column for this lane

    v8f acc = {};
    const float* arow_p = A + (size_t)arow * K;
    for (int k0 = 0; k0 < K; k0 += 4) {
        const int ka = k0 + (half << 1);
        v2f a, b;
        a.x = arow_p[ka];
        a.y = arow_p[ka + 1];
        b.x = W[(size_t)ka * O + bcol];
        b.y = W[(size_t)(ka + 1) * O + bcol];
        acc = __builtin_amdgcn_wmma_f32_16x16x4_f32(
            /*neg_a=*/false, a, /*neg_b=*/false, b,
            /*c_mod=*/(short)0, acc, /*reuse_a=*/false, /*reuse_b=*/false);
    }
#pragma unroll
    for (int i = 0; i < 8; ++i) {
        C[(size_t)(m0 + i + (half << 3)) * O + bcol] = acc[i];
    }
}

// ---------------- aggregation ----------------
// out[i,f] = xw[i,f]*dinv[i]^2 + bias[f]   (self-loop term + bias, initializes out)
__global__ void agg_init_kernel(const float* __restrict__ xw, const float* __restrict__ dinv,
                                const float* __restrict__ bias, float* __restrict__ out,
                                int n, int F) {
    int tid = blockIdx.x * blockDim.x + threadIdx.x;
    if (tid >= n * F) return;
    int i = tid / F, f = tid - i * F;
    float di = dinv[i];
    out[tid] = xw[tid] * di * di + bias[f];
}

// out[dst] += xw[src] * norm[e]   (float4 chunks, hardware f32 atomics)
__global__ void agg_edges_kernel(const float* __restrict__ xw, const int* __restrict__ src,
                                 const int* __restrict__ dst, const float* __restrict__ norm,
                                 float* __restrict__ out, int E, int F) {
    const int cpe = F >> 2;                       // float4 chunks per edge
    long long tid = (long long)blockIdx.x * blockDim.x + threadIdx.x;
    if (tid >= (long long)E * cpe) return;
    int e = (int)(tid / cpe);
    int c = (int)(tid - (long long)e * cpe) << 2;
    int s = src[e], d = dst[e];
    float nm = norm[e];
    const float4 v = *(const float4*)(xw + (size_t)s * F + c);
    float* o = out + (size_t)d * F + c;
    unsafeAtomicAdd(o + 0, v.x * nm);
    unsafeAtomicAdd(o + 1, v.y * nm);
    unsafeAtomicAdd(o + 2, v.z * nm);
    unsafeAtomicAdd(o + 3, v.w * nm);
}

// ---------------- batchnorm ----------------
// blockDim.x == F; each block reduces a row slice, coalesced across threads.
__global__ void bn_stats_kernel(const float* __restrict__ x, float* __restrict__ gsum,
                                float* __restrict__ gsumsq, int n, int F,
                                int rowsPerBlock, int reluFirst) {
    int f  = threadIdx.x;
    int r0 = blockIdx.x * rowsPerBlock;
    int r1 = min(r0 + rowsPerBlock, n);
    float s = 0.f, ss = 0.f;
    for (int r = r0; r < r1; ++r) {
        float v = x[(size_t)r * F + f];
        if (reluFirst) v = fmaxf(v, 0.f);
        s += v; ss += v * v;
    }
    unsafeAtomicAdd(&gsum[f], s);
    unsafeAtomicAdd(&gsumsq[f], ss);
}

__global__ void bn_finalize_kernel(const float* __restrict__ gsum, const float* __restrict__ gsumsq,
                                   const float* __restrict__ gamma, const float* __restrict__ beta,
                                   float* __restrict__ scale, float* __restrict__ shift,
                                   int n, int F) {
    int f = blockIdx.x * blockDim.x + threadIdx.x;
    if (f >= F) return;
    float inv_n = 1.0f / (float)n;
    float mean = gsum[f] * inv_n;
    float var  = gsumsq[f] * inv_n - mean * mean;   // biased variance (jnp.var)
    float sc   = gamma[f] * rsqrtf(var + 1e-5f);
    scale[f] = sc;
    shift[f] = beta[f] - mean * sc;
}

// mode 0: y = relu(x*scale+shift)   (mu branch: BN then ReLU)
// mode 1: y = relu(x)*scale+shift   (logstd branch: ReLU then BN)
__global__ void bn_apply_kernel(const float* __restrict__ x, const float* __restrict__ scale,
                                const float* __restrict__ shift, float* __restrict__ y,
                                int total, int F, int mode) {
    int tid = blockIdx.x * blockDim.x + threadIdx.x;
    if (tid >= total) return;
    int f = tid % F;
    float v = x[tid];
    if (mode == 0) y[tid] = fmaxf(v * scale[f] + shift[f], 0.f);
    else           y[tid] = fmaxf(v, 0.f) * scale[f] + shift[f];
}

// ---------------- epilogue ----------------
__global__ void clamp_kernel(float* __restrict__ x, int total, float hi) {
    int tid = blockIdx.x * blockDim.x + threadIdx.x;
    if (tid >= total) return;
    x[tid] = fminf(x[tid], hi);
}

__global__ void decoder_kernel(const float* __restrict__ mu, const int* __restrict__ src,
                               const int* __restrict__ dst, float* __restrict__ probs, int E) {
    int e = blockIdx.x * blockDim.x + threadIdx.x;
    if (e >= E) return;
    const float4* a = (const float4*)(mu + (size_t)src[e] * 32);
    const float4* b = (const float4*)(mu + (size_t)dst[e] * 32);
    float dot = 0.f;
#pragma unroll
    for (int i = 0; i < 8; ++i) {
        float4 u = a[i], v = b[i];
        dot += u.x * v.x + u.y * v.y + u.z * v.z + u.w * v.w;
    }
    probs[e] = 1.0f / (1.0f + __expf(-dot));
}

// ---------------------------------------------------------------------------
static inline int cdiv(long long a, long long b) { return (int)((a + b - 1) / b); }

extern "C" void kernel_launch(void* const* d_in, const int* in_sizes, int n_in,
                              void* d_out, int out_size, void* d_ws, size_t ws_size,
                              hipStream_t stream) {
    const int D_IN = 64, H1 = 64, H2 = 128, Z = 32;
    const int N = in_sizes[0] / D_IN;         // 50000
    const int E = in_sizes[1] / 2;            // 800000

    const float* x   = (const float*)d_in[0];
    const int*   src = (const int*)d_in[1];
    const int*   dst = src + E;

    const float* Wm1 = (const float*)d_in[2];  const float* bm1 = (const float*)d_in[3];
    const float* gm1 = (const float*)d_in[4];  const float* hm1 = (const float*)d_in[5];
    const float* Wm2 = (const float*)d_in[6];  const float* bm2 = (const float*)d_in[7];
    const float* gm2 = (const float*)d_in[8];  const float* hm2 = (const float*)d_in[9];
    const float* Wm3 = (const float*)d_in[10]; const float* bm3 = (const float*)d_in[11];
    const float* Wl1 = (const float*)d_in[12]; const float* bl1 = (const float*)d_in[13];
    const float* gl1 = (const float*)d_in[14]; const float* hl1 = (const float*)d_in[15];
    const float* Wl2 = (const float*)d_in[16]; const float* bl2 = (const float*)d_in[17];
    const float* gl2 = (const float*)d_in[18]; const float* hl2 = (const float*)d_in[19];
    const float* Wl3 = (const float*)d_in[20]; const float* bl3 = (const float*)d_in[21];

    // outputs: [edge_probs E][mu N*Z][logstd N*Z]
    float* probs  = (float*)d_out;
    float* mu     = probs + E;
    float* logstd = mu + (size_t)N * Z;

    // workspace layout (all 256B aligned)
    char* ws = (char*)d_ws;
    size_t off = 0;
    auto take = [&](size_t bytes) { char* p = ws + off; off += (bytes + 255) & ~(size_t)255; return p; };
    float* dinv   = (float*)take((size_t)N * 4);
    float* norm   = (float*)take((size_t)E * 4);
    float* bufA   = (float*)take((size_t)N * H2 * 4);
    float* bufB   = (float*)take((size_t)N * H2 * 4);
    float* gsum   = (float*)take(128 * 4);
    float* gsumsq = (float*)take(128 * 4);
    float* scale  = (float*)take(128 * 4);
    float* shift  = (float*)take(128 * 4);
    (void)ws_size; (void)n_in; (void)out_size;

    // ---- graph normalization ----
    hipMemsetAsync(dinv, 0, (size_t)N * 4, stream);
    deg_accum_kernel<<<cdiv(E, TPB), TPB, 0, stream>>>(dst, dinv, E);
    deg_to_dinv_kernel<<<cdiv(N, TPB), TPB, 0, stream>>>(dinv, N);
    edge_norm_kernel<<<cdiv(E, TPB), TPB, 0, stream>>>(src, dst, dinv, norm, E);

    const int ROWS = 512;
    const int statBlocks = cdiv(N, ROWS);

    auto gemm = [&](const float* A, const float* W, float* C, int K, int O) {
        int tiles = (N >> 4) * (O >> 4);
        gemm_wmma_kernel<<<cdiv(tiles, 4), 128, 0, stream>>>(A, W, C, N, K, O);
    };
    auto aggregate = [&](const float* xw, const float* bias, float* out, int F) {
        agg_init_kernel<<<cdiv((long long)N * F, TPB), TPB, 0, stream>>>(xw, dinv, bias, out, N, F);
        agg_edges_kernel<<<cdiv((long long)E * (F >> 2), TPB), TPB, 0, stream>>>(
            xw, src, dst, norm, out, E, F);
    };
    auto batchnorm = [&](const float* in, float* out, const float* gamma, const float* beta,
                         int F, int mode) {
        hipMemsetAsync(gsum, 0, 128 * 4, stream);
        hipMemsetAsync(gsumsq, 0, 128 * 4, stream);
        bn_stats_kernel<<<statBlocks, F, 0, stream>>>(in, gsum, gsumsq, N, F, ROWS, mode);
        bn_finalize_kernel<<<1, F, 0, stream>>>(gsum, gsumsq, gamma, beta, scale, shift, N, F);
        bn_apply_kernel<<<cdiv((long long)N * F, TPB), TPB, 0, stream>>>(
            in, scale, shift, out, N * F, F, mode);
    };

    auto branch = [&](const float* W1, const float* b1, const float* g1, const float* h1,
                      const float* W2, const float* b2, const float* g2, const float* h2,
                      const float* W3, const float* b3, float* outZ, int mode) {
        // layer 1: x[N,64] @ W1[64,64] -> bufB; aggregate -> bufA; BN/ReLU -> bufB
        gemm(x, W1, bufB, D_IN, H1);
        aggregate(bufB, b1, bufA, H1);
        batchnorm(bufA, bufB, g1, h1, H1, mode);
        // layer 2: bufB[N,64] @ W2[64,128] -> bufA; aggregate -> bufB; BN/ReLU -> bufA
        gemm(bufB, W2, bufA, H1, H2);
        aggregate(bufA, b2, bufB, H2);
        batchnorm(bufB, bufA, g2, h2, H2, mode);
        // layer 3: bufA[N,128] @ W3[128,32] -> bufB; aggregate -> outZ
        gemm(bufA, W3, bufB, H2, Z);
        aggregate(bufB, b3, outZ, Z);
    };

    branch(Wm1, bm1, gm1, hm1, Wm2, bm2, gm2, hm2, Wm3, bm3, mu, 0);
    branch(Wl1, bl1, gl1, hl1, Wl2, bl2, gl2, hl2, Wl3, bl3, logstd, 1);

    clamp_kernel<<<cdiv((long long)N * Z, TPB), TPB, 0, stream>>>(logstd, N * Z, 10.0f);
    decoder_kernel<<<cdiv(E, TPB), TPB, 0, stream>>>(mu, src, dst, probs, E);
}